// CombineSF_30623116821153
// MI455X (gfx1250) — compile-verified
//
#include <hip/hip_runtime.h>
#include <stdint.h>

// ---------------------------------------------------------------------------
// CombineSF (MoE combine / segment-sum) for gfx1250.
// Strategy: build an inverse index (per-tag linked list) with cheap u32
// atomics, then one wave32 per output row walks its chain, staging each 1KB
// source row into LDS via CDNA5 async global->LDS loads (ASYNCcnt,
// double-buffered), accumulates in VGPRs, and streams the row out once with
// b128 stores. Pure-streaming traffic ~0.78 GB => ~34us at 23.3 TB/s.
// ---------------------------------------------------------------------------

#define DCOLS            256        // row width in floats (reference D)
#define ROW_BYTES        1024       // DCOLS * 4
#define WAVES_PER_BLOCK  8
#define INVALID          0xFFFFFFFFu

// Issue an async copy of one 1KB row (global -> LDS). Two b128 ops per lane
// (32 lanes x 16B = 512B each). Per CDNA5 ISA, INST_OFFSET is
// added to BOTH the LDS and the global address, so one base pair suffices.
__device__ __forceinline__ void async_row_to_lds(const float* rowBase,
                                                 uint32_t ldsByteAddr,
                                                 int lane) {
    uint64_t g = (uint64_t)(uintptr_t)rowBase + (uint32_t)(lane * 16);
    uint32_t l = ldsByteAddr + (uint32_t)(lane * 16);
    asm volatile("global_load_async_to_lds_b128 %0, %1, off"
                 :: "v"(l), "v"(g) : "memory");
    asm volatile("global_load_async_to_lds_b128 %0, %1, off offset:512"
                 :: "v"(l), "v"(g) : "memory");
}

__device__ __forceinline__ const float* row_ptr(uint32_t r,
                                                const float* d0, const float* d1,
                                                const float* d2, const float* d3,
                                                uint32_t nPer) {
    uint32_t f = r / nPer;
    uint32_t k = r - f * nPer;
    const float* b = d0;
    if (f == 1) b = d1;
    else if (f == 2) b = d2;
    else if (f == 3) b = d3;
    return b + (size_t)k * DCOLS;
}

// --- Phase 1: head[] = -1 ---------------------------------------------------
__global__ void combine_init_heads(uint32_t* __restrict__ head, int n) {
    int i = blockIdx.x * blockDim.x + threadIdx.x;
    int n4 = n >> 2;
    if (i < n4) {
        reinterpret_cast<uint4*>(head)[i] = make_uint4(INVALID, INVALID, INVALID, INVALID);
    }
    int rem = n & 3;
    if (i < rem) head[(n4 << 2) + i] = INVALID;
}

// --- Phase 2: per-tag linked lists via atomicExch ---------------------------
__global__ void combine_build_lists(const int* __restrict__ t0, const int* __restrict__ t1,
                                    const int* __restrict__ t2, const int* __restrict__ t3,
                                    uint32_t* __restrict__ head, uint32_t* __restrict__ nxt,
                                    int nPer) {
    int i = blockIdx.x * blockDim.x + threadIdx.x;
    int total = 4 * nPer;
    if (i >= total) return;
    int f = i / nPer;
    int r = i - f * nPer;
    const int* tp = (f == 0) ? t0 : (f == 1) ? t1 : (f == 2) ? t2 : t3;
    uint32_t tag = (uint32_t)tp[r];
    uint32_t prev = atomicExch(&head[tag], (uint32_t)i);
    nxt[i] = prev;
}

// --- Phase 3: one wave per output row; async-LDS double-buffered gather -----
__global__ void __launch_bounds__(32 * WAVES_PER_BLOCK)
combine_gather(const float* __restrict__ d0, const float* __restrict__ d1,
               const float* __restrict__ d2, const float* __restrict__ d3,
               const uint32_t* __restrict__ head, const uint32_t* __restrict__ nxt,
               float* __restrict__ out, int load, int nPer) {
    __shared__ float lds[WAVES_PER_BLOCK][2][DCOLS];   // 16 KB per block

    const int lane = threadIdx.x & 31;
    const int w    = threadIdx.x >> 5;
    const int gw   = blockIdx.x * WAVES_PER_BLOCK + w;
    const int nw   = gridDim.x * WAVES_PER_BLOCK;
    const uint32_t uPer = (uint32_t)nPer;

    // Wave-relative LDS byte offset: low 32 bits of the flat shared address
    // (CDNA5 ISA: flat->LDS uses addr[31:0] relative to the wave's LDS).
    const uint32_t lds0 = (uint32_t)(uintptr_t)(&lds[w][0][0]);
    const uint32_t lds1 = lds0 + ROW_BYTES;

    for (int t = gw; t < load; t += nw) {
        float4 acc0 = make_float4(0.f, 0.f, 0.f, 0.f);
        float4 acc1 = make_float4(0.f, 0.f, 0.f, 0.f);

        uint32_t cur = head[t];                // wave-uniform
        if (cur != INVALID) {
            int buf = 0;
            async_row_to_lds(row_ptr(cur, d0, d1, d2, d3, uPer), lds0, lane);
            uint32_t nx = nxt[cur];
            for (;;) {
                // Buffer `buf` is the oldest outstanding async pair: drain it.
                asm volatile("s_wait_asynccnt 0" ::: "memory");
                uint32_t nx2 = INVALID;
                if (nx != INVALID) {
                    // Prefetch next chain element into the other buffer while
                    // we accumulate the current one out of LDS.
                    async_row_to_lds(row_ptr(nx, d0, d1, d2, d3, uPer),
                                     buf ? lds0 : lds1, lane);
                    nx2 = nxt[nx];
                }
                const float4* p =
                    reinterpret_cast<const float4*>(&lds[w][buf][lane * 8]);
                float4 v0 = p[0];
                float4 v1 = p[1];
                acc0.x += v0.x; acc0.y += v0.y; acc0.z += v0.z; acc0.w += v0.w;
                acc1.x += v1.x; acc1.y += v1.y; acc1.z += v1.z; acc1.w += v1.w;
                if (nx == INVALID) break;
                buf ^= 1;
                nx = nx2;
            }
        }
        // Single streaming write of the finished row (also zero-fills empty
        // rows, clearing the harness's 0xAA poison).
        float4* o = reinterpret_cast<float4*>(out + (size_t)t * DCOLS + lane * 8);
        o[0] = acc0;
        o[1] = acc1;
    }
}

// ---------------------------------------------------------------------------
extern "C" void kernel_launch(void* const* d_in, const int* in_sizes, int n_in,
                              void* d_out, int out_size, void* d_ws, size_t ws_size,
                              hipStream_t stream) {
    const int D_   = DCOLS;
    const int nPer = in_sizes[0] / D_;      // 131072 rows per flow
    const int load = out_size / D_;         // 262144 output rows

    // setup_inputs() insertion order: data0,tags0,data1,tags1,data2,tags2,data3,tags3,load
    const float* d0 = (const float*)d_in[0];
    const int*   t0 = (const int*)  d_in[1];
    const float* d1 = (const float*)d_in[2];
    const int*   t1 = (const int*)  d_in[3];
    const float* d2 = (const float*)d_in[4];
    const int*   t2 = (const int*)  d_in[5];
    const float* d3 = (const float*)d_in[6];
    const int*   t3 = (const int*)  d_in[7];

    uint32_t* head = (uint32_t*)d_ws;       // load u32    (1 MB)
    uint32_t* nxt  = head + load;           // 4*nPer u32  (2 MB)

    {   // Phase 1
        int work = (load >> 2) + 4;
        int th = 256, bl = (work + th - 1) / th;
        combine_init_heads<<<bl, th, 0, stream>>>(head, load);
    }
    {   // Phase 2
        int total = 4 * nPer;
        int th = 256, bl = (total + th - 1) / th;
        combine_build_lists<<<bl, th, 0, stream>>>(t0, t1, t2, t3, head, nxt, nPer);
    }
    {   // Phase 3
        int th = 32 * WAVES_PER_BLOCK;
        int bl = (load + WAVES_PER_BLOCK - 1) / WAVES_PER_BLOCK;
        combine_gather<<<bl, th, 0, stream>>>(d0, d1, d2, d3, head, nxt,
                                              (float*)d_out, load, nPer);
    }
}